// LSTMAttentionDecoder_29240137351302
// MI455X (gfx1250) — compile-verified
//
#include <hip/hip_runtime.h>
#include <math.h>

// ---------------- problem constants (match reference) ----------------
#define B_    32
#define T_    128
#define S_    256
#define H_    512
#define IN_   512
#define V_    16000
#define HEADS_ 2
#define G4_   2048          // 4*H
#define ZW_   2048          // zin row width: [ctx0(512), ctx1(512), x_t(512), h(512)]
#define EPS_  1e-5f

typedef float v2f __attribute__((ext_vector_type(2)));
typedef float v8f __attribute__((ext_vector_type(8)));

// ---------------- block reductions (256 threads) ----------------
__device__ __forceinline__ float blk_sum256(float v, float* sb) {
    const int t = threadIdx.x;
    sb[t] = v; __syncthreads();
#pragma unroll
    for (int s = 128; s > 0; s >>= 1) {
        if (t < s) sb[t] += sb[t + s];
        __syncthreads();
    }
    float r = sb[0]; __syncthreads();
    return r;
}

__device__ __forceinline__ float blk_max256(float v, float* sb) {
    const int t = threadIdx.x;
    sb[t] = v; __syncthreads();
#pragma unroll
    for (int s = 128; s > 0; s >>= 1) {
        if (t < s) sb[t] = fmaxf(sb[t], sb[t + s]);
        __syncthreads();
    }
    float r = sb[0]; __syncthreads();
    return r;
}

__device__ __forceinline__ float sigm_(float x) { return 1.0f / (1.0f + expf(-x)); }

// ---------------- fp32 WMMA GEMM:  Y[M,N] = X[M,K](ld=ldx) @ W[N,K]^T (+bias) ----------------
// wave32; each wave computes a 32x16 slab (two 16x16 C tiles sharing the B fragment).
// A (16x4 f32): lane -> row M=lane%16; VGPR v holds K = 2*(lane/16)+v  => contiguous float2.
// B (4x16 f32): lane -> col N=lane%16; VGPR v holds K = 2*(lane/16)+v  => contiguous float2 of W row.
// C/D (16x16 f32): VGPR v -> row v + 8*(lane/16), col lane%16.
__global__ __launch_bounds__(256)
void gemm_f32_wmma(const float* __restrict__ X, int ldx,
                   const float* __restrict__ W,
                   const float* __restrict__ bias,
                   float* __restrict__ Y, int ldy,
                   int M, int N, int K) {
    const int wave = (blockIdx.x * blockDim.x + threadIdx.x) >> 5;
    const int lane = threadIdx.x & 31;
    const int mPairs = M >> 5;                 // 32 rows per wave
    const int nTiles = N >> 4;
    const int ni = wave / mPairs;
    const int mp = wave % mPairs;
    if (ni >= nTiles) return;                  // whole-wave uniform exit: EXEC stays all-1s

    const int hi = lane >> 4;                  // lane half selects K sub-pair
    const int lr = lane & 15;
    const int nr = ni * 16 + lr;               // W row == output column

    const float* xrow0 = X + (size_t)(mp * 32 + lr) * ldx;
    const float* xrow1 = xrow0 + (size_t)16 * ldx;
    const float* wrow  = W + (size_t)nr * K;

    v8f c0 = {0.f, 0.f, 0.f, 0.f, 0.f, 0.f, 0.f, 0.f};
    v8f c1 = {0.f, 0.f, 0.f, 0.f, 0.f, 0.f, 0.f, 0.f};

#pragma unroll 8
    for (int kk = 0; kk < K; kk += 4) {
        const int kb = kk + 2 * hi;
        v2f bf = *reinterpret_cast<const v2f*>(wrow  + kb);
        v2f a0 = *reinterpret_cast<const v2f*>(xrow0 + kb);
        v2f a1 = *reinterpret_cast<const v2f*>(xrow1 + kb);
        c0 = __builtin_amdgcn_wmma_f32_16x16x4_f32(false, a0, false, bf, (short)0, c0, false, false);
        c1 = __builtin_amdgcn_wmma_f32_16x16x4_f32(false, a1, false, bf, (short)0, c1, false, false);
    }

    const float bv = bias ? bias[nr] : 0.0f;
    {
        float* y = Y + (size_t)(mp * 32 + 8 * hi) * ldy + nr;
#pragma unroll
        for (int v = 0; v < 8; ++v) y[(size_t)v * ldy] = c0[v] + bv;
    }
    {
        float* y = Y + (size_t)(mp * 32 + 16 + 8 * hi) * ldy + nr;
#pragma unroll
        for (int v = 0; v < 8; ++v) y[(size_t)v * ldy] = c1[v] + bv;
    }
}

static inline void launch_gemm(const float* X, int ldx, const float* W, const float* bias,
                               float* Y, int ldy, int M, int N, int K, hipStream_t s) {
    const int waves = (M >> 5) * (N >> 4);
    const int blocks = (waves * 32 + 255) / 256;
    gemm_f32_wmma<<<blocks, 256, 0, s>>>(X, ldx, W, bias, Y, ldy, M, N, K);
}

// ---------------- LayerNorm of preconditioned stacks (in-place), per-head gain/bias ----------------
__global__ __launch_bounds__(256)
void ln_precon_kernel(float* __restrict__ precon, const float* __restrict__ g,
                      const float* __restrict__ be) {
    __shared__ float sb[256];
    const int row = blockIdx.x;                 // 0 .. HEADS*B*S-1
    const int n = row / (B_ * S_);
    float* x = precon + (size_t)row * H_;
    const int t = threadIdx.x;
    float v0 = x[t], v1 = x[t + 256];
    float m = blk_sum256(v0 + v1, sb) * (1.0f / H_);
    float d0 = v0 - m, d1 = v1 - m;
    float inv = rsqrtf(blk_sum256(d0 * d0 + d1 * d1, sb) * (1.0f / H_) + EPS_);
    x[t]       = d0 * inv * g[n * H_ + t]       + be[n * H_ + t];
    x[t + 256] = d1 * inv * g[n * H_ + t + 256] + be[n * H_ + t + 256];
}

// ---------------- init h,c ----------------
__global__ void init_state_kernel(const float* __restrict__ h0, const float* __restrict__ c0,
                                  float* __restrict__ h, float* __restrict__ c) {
    int idx = blockIdx.x * blockDim.x + threadIdx.x;
    if (idx < B_ * H_) { h[idx] = h0[idx & (H_ - 1)]; c[idx] = c0[idx & (H_ - 1)]; }
}

// ---------------- attention + context; writes ctx into zin[:, n*512 .. ] ----------------
__global__ __launch_bounds__(256)
void attn_ctx_kernel(const float* __restrict__ precon,
                     const float* __restrict__ stack0, const float* __restrict__ stack1,
                     const int* __restrict__ mask0, const int* __restrict__ mask1,
                     const float* __restrict__ hbuf, float* __restrict__ zin) {
    __shared__ float sh[H_];
    __shared__ float sw[S_];
    __shared__ float sb[256];
    const int n = blockIdx.x / B_;
    const int b = blockIdx.x % B_;
    const int t = threadIdx.x;                  // t == s index (S_ == 256)

    sh[t]       = hbuf[b * H_ + t];
    sh[t + 256] = hbuf[b * H_ + t + 256];
    __syncthreads();

    // logits[n,b,s] = dot(precon[n,b,s,:], h[b,:])
    const float*  pr  = precon + (((size_t)n * B_ + b) * S_ + t) * H_;
    const float4* pr4 = reinterpret_cast<const float4*>(pr);
    float acc = 0.f;
#pragma unroll 4
    for (int i = 0; i < H_ / 4; ++i) {
        float4 p = pr4[i];
        acc += p.x * sh[4 * i] + p.y * sh[4 * i + 1] + p.z * sh[4 * i + 2] + p.w * sh[4 * i + 3];
    }
    const int mk = (n == 0 ? mask0 : mask1)[b * S_ + t];
    float logit = (mk == 0) ? -1e10f : acc;

    float mx = blk_max256(logit, sb);
    float e  = expf(logit - mx);
    float sm = blk_sum256(e, sb);
    sw[t] = e / sm;
    __syncthreads();

    // ctx[h] = sum_s w[s] * stacks[n,b,s,h]  (coalesced column sweep)
    const float* stk = (n == 0 ? stack0 : stack1) + (size_t)b * S_ * H_;
    float a0 = 0.f, a1 = 0.f;
    for (int s = 0; s < S_; ++s) {
        float w = sw[s];
        a0 += w * stk[s * H_ + t];
        a1 += w * stk[s * H_ + t + 256];
    }
    zin[b * ZW_ + n * H_ + t]       = a0;
    zin[b * ZW_ + n * H_ + t + 256] = a1;
}

// ---------------- assemble x_t and h into zin[:, 1024..2047] ----------------
__global__ __launch_bounds__(256)
void assemble_kernel(const float* __restrict__ embeds, const float* __restrict__ hbuf,
                     float* __restrict__ zin, int t) {
    const int b = blockIdx.x, j = threadIdx.x;
    const float* x = embeds + ((size_t)b * T_ + t) * IN_;
    zin[b * ZW_ + 1024 + j]       = x[j];
    zin[b * ZW_ + 1024 + j + 256] = x[j + 256];
    zin[b * ZW_ + 1536 + j]       = hbuf[b * H_ + j];
    zin[b * ZW_ + 1536 + j + 256] = hbuf[b * H_ + j + 256];
}

// ---------------- Y[row] = tanh(LN(X[row]) * g + b); row width 512 ----------------
__global__ __launch_bounds__(256)
void ln_tanh_kernel(const float* __restrict__ X, const float* __restrict__ g,
                    const float* __restrict__ be, float* __restrict__ Y, long long ldy) {
    __shared__ float sb[256];
    const int b = blockIdx.x, t = threadIdx.x;
    const float* x = X + b * H_;
    float v0 = x[t], v1 = x[t + 256];
    float m = blk_sum256(v0 + v1, sb) * (1.0f / H_);
    float d0 = v0 - m, d1 = v1 - m;
    float inv = rsqrtf(blk_sum256(d0 * d0 + d1 * d1, sb) * (1.0f / H_) + EPS_);
    float* y = Y + (size_t)b * ldy;
    y[t]       = tanhf(d0 * inv * g[t]       + be[t]);
    y[t + 256] = tanhf(d1 * inv * g[t + 256] + be[t + 256]);
}

// ---------------- copy gate: sigmoid(tanh(LN(c1out)) . W_c2 + b_c2) ----------------
__global__ __launch_bounds__(256)
void copy_gate_kernel(const float* __restrict__ X, const float* __restrict__ g,
                      const float* __restrict__ be, const float* __restrict__ Wc2,
                      const float* __restrict__ bc2, float* __restrict__ outCp, int t) {
    __shared__ float sb[256];
    const int b = blockIdx.x, j = threadIdx.x;
    const float* x = X + b * H_;
    float v0 = x[j], v1 = x[j + 256];
    float m = blk_sum256(v0 + v1, sb) * (1.0f / H_);
    float d0 = v0 - m, d1 = v1 - m;
    float inv = rsqrtf(blk_sum256(d0 * d0 + d1 * d1, sb) * (1.0f / H_) + EPS_);
    float h0 = tanhf(d0 * inv * g[j]       + be[j]);
    float h1 = tanhf(d1 * inv * g[j + 256] + be[j + 256]);
    float s  = blk_sum256(h0 * Wc2[j] + h1 * Wc2[j + 256], sb);
    if (j == 0) outCp[b * T_ + t] = sigm_(s + bc2[0]);
}

// ---------------- LSTM cell: gates = LN(ih)+LN(hh); update c,h; mirror h into zin ----------------
__global__ __launch_bounds__(256)
void lstm_kernel(const float* __restrict__ ih, const float* __restrict__ hh,
                 const float* __restrict__ gih, const float* __restrict__ bih,
                 const float* __restrict__ ghh, const float* __restrict__ bhh,
                 float* __restrict__ hbuf, float* __restrict__ cbuf, float* __restrict__ zin) {
    __shared__ float sb[256];
    const int b = blockIdx.x, t = threadIdx.x;
    const float* xr = ih + (size_t)b * G4_;
    const float* yr = hh + (size_t)b * G4_;

    float xv[8], yv[8], xs = 0.f, ys = 0.f;
#pragma unroll
    for (int i = 0; i < 8; ++i) {
        xv[i] = xr[t + 256 * i]; yv[i] = yr[t + 256 * i];
        xs += xv[i]; ys += yv[i];
    }
    float mx = blk_sum256(xs, sb) * (1.0f / G4_);
    float my = blk_sum256(ys, sb) * (1.0f / G4_);
    float qx = 0.f, qy = 0.f;
#pragma unroll
    for (int i = 0; i < 8; ++i) {
        float dx = xv[i] - mx, dy = yv[i] - my;
        qx += dx * dx; qy += dy * dy;
    }
    float ivx = rsqrtf(blk_sum256(qx, sb) * (1.0f / G4_) + EPS_);
    float ivy = rsqrtf(blk_sum256(qy, sb) * (1.0f / G4_) + EPS_);

    for (int j = t; j < H_; j += 256) {
        const int ji = j, jf = j + 512, jg = j + 1024, jo = j + 1536;
        float gi = (xr[ji] - mx) * ivx * gih[ji] + bih[ji] + (yr[ji] - my) * ivy * ghh[ji] + bhh[ji];
        float gf = (xr[jf] - mx) * ivx * gih[jf] + bih[jf] + (yr[jf] - my) * ivy * ghh[jf] + bhh[jf];
        float gg = (xr[jg] - mx) * ivx * gih[jg] + bih[jg] + (yr[jg] - my) * ivy * ghh[jg] + bhh[jg];
        float go = (xr[jo] - mx) * ivx * gih[jo] + bih[jo] + (yr[jo] - my) * ivy * ghh[jo] + bhh[jo];
        float cn = sigm_(gf) * cbuf[b * H_ + j] + sigm_(gi) * tanhf(gg);
        float hn = sigm_(go) * tanhf(cn);
        cbuf[b * H_ + j] = cn;
        hbuf[b * H_ + j] = hn;
        zin[b * ZW_ + 1024 + j] = hn;   // mh input = [ctx_flat, h_new]
    }
}

// ---------------- host orchestration ----------------
extern "C" void kernel_launch(void* const* d_in, const int* in_sizes, int n_in,
                              void* d_out, int out_size, void* d_ws, size_t ws_size,
                              hipStream_t stream) {
    (void)in_sizes; (void)n_in; (void)out_size; (void)ws_size;

    const float* embeds = (const float*)d_in[0];
    const float* stack0 = (const float*)d_in[1];
    const float* stack1 = (const float*)d_in[2];
    const int*   mask0  = (const int*)  d_in[3];
    const int*   mask1  = (const int*)  d_in[4];
    const float* att_W  = (const float*)d_in[5];
    const float* att_g  = (const float*)d_in[6];
    const float* att_b  = (const float*)d_in[7];
    const float* W_ih   = (const float*)d_in[8];
    const float* W_hh   = (const float*)d_in[9];
    const float* b_ih   = (const float*)d_in[10];
    const float* b_hh   = (const float*)d_in[11];
    const float* ln_ih_g = (const float*)d_in[12];
    const float* ln_ih_b = (const float*)d_in[13];
    const float* ln_hh_g = (const float*)d_in[14];
    const float* ln_hh_b = (const float*)d_in[15];
    const float* W_mi   = (const float*)d_in[16];
    const float* b_mi   = (const float*)d_in[17];
    const float* mi_g   = (const float*)d_in[18];
    const float* mi_b   = (const float*)d_in[19];
    const float* W_mh   = (const float*)d_in[20];
    const float* b_mh   = (const float*)d_in[21];
    const float* mh_g   = (const float*)d_in[22];
    const float* mh_b   = (const float*)d_in[23];
    const float* W_c1   = (const float*)d_in[24];
    const float* b_c1   = (const float*)d_in[25];
    const float* c_g    = (const float*)d_in[26];
    const float* c_b    = (const float*)d_in[27];
    const float* W_c2   = (const float*)d_in[28];
    const float* b_c2   = (const float*)d_in[29];
    const float* W_proj = (const float*)d_in[30];
    const float* b_proj = (const float*)d_in[31];
    const float* W_log  = (const float*)d_in[32];
    const float* b_log  = (const float*)d_in[33];
    const float* h0     = (const float*)d_in[34];
    const float* c0     = (const float*)d_in[35];

    float* out   = (float*)d_out;
    float* outCp = out + (size_t)B_ * T_ * V_;      // copy_probs [B,T,1]

    // workspace carve (floats); total ~12.9M floats (~51.5 MB)
    float* ws = (float*)d_ws;
    size_t off = 0;
    auto carve = [&](size_t n) { float* p = ws + off; off += n; return p; };
    float* precon = carve((size_t)HEADS_ * B_ * S_ * H_);
    float* hbuf   = carve((size_t)B_ * H_);
    float* cbuf   = carve((size_t)B_ * H_);
    float* zin    = carve((size_t)B_ * ZW_);
    float* c1out  = carve((size_t)B_ * H_);
    float* miout  = carve((size_t)B_ * H_);
    float* newin  = carve((size_t)B_ * H_);
    float* ihout  = carve((size_t)B_ * G4_);
    float* hhout  = carve((size_t)B_ * G4_);
    float* mhout  = carve((size_t)B_ * H_);
    float* outs   = carve((size_t)B_ * T_ * H_);
    float* proj   = carve((size_t)B_ * T_ * H_);

    // ---- Phase A: precondition stacks (per-head GEMM + LayerNorm), init state ----
    for (int n = 0; n < HEADS_; ++n) {
        const float* stk = (n == 0) ? stack0 : stack1;
        launch_gemm(stk, H_, att_W + (size_t)n * H_ * H_, nullptr,
                    precon + (size_t)n * B_ * S_ * H_, H_, B_ * S_, H_, H_, stream);
    }
    ln_precon_kernel<<<HEADS_ * B_ * S_, 256, 0, stream>>>(precon, att_g, att_b);
    init_state_kernel<<<(B_ * H_ + 255) / 256, 256, 0, stream>>>(h0, c0, hbuf, cbuf);

    // ---- Phase B: sequential scan over T ----
    for (int t = 0; t < T_; ++t) {
        attn_ctx_kernel<<<HEADS_ * B_, 256, 0, stream>>>(precon, stack0, stack1,
                                                         mask0, mask1, hbuf, zin);
        assemble_kernel<<<B_, 256, 0, stream>>>(embeds, hbuf, zin, t);
        // copy-gate MLP (input = zin[:,0:2048]) and merge-input (input = zin[:,0:1536])
        launch_gemm(zin, ZW_, W_c1, b_c1, c1out, H_, B_, H_, 2048, stream);
        launch_gemm(zin, ZW_, W_mi, b_mi, miout, H_, B_, H_, 1536, stream);
        copy_gate_kernel<<<B_, 256, 0, stream>>>(c1out, c_g, c_b, W_c2, b_c2, outCp, t);
        ln_tanh_kernel<<<B_, 256, 0, stream>>>(miout, mi_g, mi_b, newin, (long long)H_);
        // gates
        launch_gemm(newin, H_, W_ih, b_ih, ihout, G4_, B_, G4_, H_, stream);
        launch_gemm(hbuf,  H_, W_hh, b_hh, hhout, G4_, B_, G4_, H_, stream);
        lstm_kernel<<<B_, 256, 0, stream>>>(ihout, hhout, ln_ih_g, ln_ih_b,
                                            ln_hh_g, ln_hh_b, hbuf, cbuf, zin);
        // out_t = tanh(LN([ctx_flat, h_new] @ W_mh^T + b_mh))  (input = zin[:,0:1536])
        launch_gemm(zin, ZW_, W_mh, b_mh, mhout, H_, B_, H_, 1536, stream);
        ln_tanh_kernel<<<B_, 256, 0, stream>>>(mhout, mh_g, mh_b,
                                               outs + (size_t)t * H_, (long long)T_ * H_);
    }

    // ---- Phase C: projection + vocab logits (written straight into d_out) ----
    launch_gemm(outs, H_, W_proj, b_proj, proj, IN_, B_ * T_, IN_, H_, stream);
    launch_gemm(proj, IN_, W_log, b_log, out, V_, B_ * T_, V_, IN_, stream);
}